// AnomalyAttention_64226940944629
// MI455X (gfx1250) — compile-verified
//
#include <hip/hip_runtime.h>
#include <math.h>

typedef __attribute__((ext_vector_type(2))) float v2f;
typedef __attribute__((ext_vector_type(4))) float v4f;
typedef __attribute__((ext_vector_type(8))) float v8f;

#define WIN  1024
#define NB   4
#define NH   8
#define EDIM 64

// Strip kernel: one workgroup handles rows [l0, l0+16) of one (b,h) attention
// matrix. 4 waves. LDS holds the 16x1024 fp32 score strip (64 KB).
__global__ __launch_bounds__(128)
void anomaly_attn_kernel(const float* __restrict__ q,
                         const float* __restrict__ k,
                         const float* __restrict__ v,
                         const float* __restrict__ sigma,
                         float* __restrict__ outV,
                         float* __restrict__ outSeries,
                         float* __restrict__ outPrior,
                         float* __restrict__ outSig) {
  extern __shared__ float sc[];          // [16][1024] scores -> series
  const int tid  = threadIdx.x;
  const int lane = tid & 31;
  const int w    = tid >> 5;             // wave id 0..3
  const int h    = blockIdx.y;
  const int b    = blockIdx.z;
  const int l0   = blockIdx.x * 16;
  const int mrow = lane & 15;            // A-matrix row index for this lane
  const int khi  = (lane >> 4) << 1;     // K sub-offset: lanes 0-15 -> 0, 16-31 -> 2

  // ================= Phase A: scores = scale * Q K^T =================
  const float scale = 0.125f;            // 1/sqrt(64)
  v2f aq[16];                            // Q strip as 16 K-chunks (A operand layout)
  {
    const float* qp = q + (((size_t)b * WIN + l0 + mrow) * NH + h) * EDIM + khi;
#pragma unroll
    for (int c2 = 0; c2 < 16; ++c2) {
      v2f t = *(const v2f*)(qp + c2 * 4);
      aq[c2] = t * scale;
    }
  }
#pragma unroll 1
  for (int tt = 0; tt < 16; ++tt) {      // wave w covers cols [256w, 256w+256)
    const int colbase = w * 256 + tt * 16;
    const float* kp = k + (((size_t)b * WIN + colbase + mrow) * NH + h) * EDIM + khi;
    v8f acc = {};
#pragma unroll
    for (int c2 = 0; c2 < 16; ++c2) {    // E = 64 = 16 chunks of K=4
      v2f bv = *(const v2f*)(kp + c2 * 4);
      acc = __builtin_amdgcn_wmma_f32_16x16x4_f32(false, aq[c2], false, bv,
                                                  (short)0, acc, false, false);
    }
#pragma unroll
    for (int r = 0; r < 8; ++r) {        // C/D layout: lanes16-31 hold rows 8..15
      int row = r + ((lane >> 4) << 3);
      sc[row * WIN + colbase + mrow] = acc[r];
    }
  }
  __syncthreads();

  // ===== Phase B+C: per-row softmax + series/prior/sig writes =====
  // wave w owns rows 4w..4w+3; wave32 shuffle reductions; b128 LDS sweeps;
  // b128 non-temporal streaming stores (output never re-read from global).
#pragma unroll 1
  for (int rr = 0; rr < 4; ++rr) {
    const int r = w * 4 + rr;
    const int l = l0 + r;
    const float lf = (float)l;

    float m = -1e30f;
    for (int g = lane; g < WIN / 4; g += 32) {
      v4f t = *(const v4f*)(&sc[r * WIN + g * 4]);
      m = fmaxf(m, fmaxf(fmaxf(t.x, t.y), fmaxf(t.z, t.w)));
    }
#pragma unroll
    for (int off = 16; off >= 1; off >>= 1) m = fmaxf(m, __shfl_xor(m, off, 32));

    float s = 0.0f;
    for (int g = lane; g < WIN / 4; g += 32) {
      v4f t = *(const v4f*)(&sc[r * WIN + g * 4]);
      s += __expf(t.x - m) + __expf(t.y - m) + __expf(t.z - m) + __expf(t.w - m);
    }
#pragma unroll
    for (int off = 16; off >= 1; off >>= 1) s += __shfl_xor(s, off, 32);
    const float inv = 1.0f / s;

    // sigma path: sig = 3^(sigmoid(5x)+1e-5) - 1
    const float x    = sigma[((size_t)b * WIN + l) * NH + h];
    const float sg   = 1.0f / (1.0f + __expf(-5.0f * x)) + 1e-5f;
    const float sigv = exp2f(sg * 1.5849625007211562f) - 1.0f;   // 3^sg = 2^(sg*log2(3))
    const float c0   = 0.3989422804014327f / sigv;                // 1/sqrt(2pi)/sig
    const float nis2 = -0.5f / (sigv * sigv);
    const v4f sig4   = {sigv, sigv, sigv, sigv};

    float* serp = outSeries + (((size_t)b * NH + h) * WIN + l) * WIN;
    float* prip = outPrior  + (((size_t)b * NH + h) * WIN + l) * WIN;
    float* sigp = outSig    + (((size_t)b * NH + h) * WIN + l) * WIN;
    for (int g = lane; g < WIN / 4; g += 32) {
      v4f t = *(const v4f*)(&sc[r * WIN + g * 4]);
      v4f e;
      e.x = __expf(t.x - m) * inv;
      e.y = __expf(t.y - m) * inv;
      e.z = __expf(t.z - m) * inv;
      e.w = __expf(t.w - m) * inv;
      *(v4f*)(&sc[r * WIN + g * 4]) = e;   // keep normalized series for Phase D
      __builtin_nontemporal_store(e, (v4f*)(serp + g * 4));

      const float cf = (float)(g * 4);
      v4f di = {lf - cf, lf - (cf + 1.0f), lf - (cf + 2.0f), lf - (cf + 3.0f)};
      v4f pr;
      pr.x = c0 * __expf(nis2 * di.x * di.x);
      pr.y = c0 * __expf(nis2 * di.y * di.y);
      pr.z = c0 * __expf(nis2 * di.z * di.z);
      pr.w = c0 * __expf(nis2 * di.w * di.w);
      __builtin_nontemporal_store(pr, (v4f*)(prip + g * 4));
      __builtin_nontemporal_store(sig4, (v4f*)(sigp + g * 4));
    }
  }
  __syncthreads();

  // ================= Phase D: V = series @ values =================
  {
    const int d0 = w * 16;               // wave w computes d-cols [16w, 16w+16)
    const float* vp = v + ((size_t)b * WIN * NH + h) * EDIM + d0 + mrow;
    v8f acc = {};
#pragma unroll 4
    for (int c2 = 0; c2 < 256; ++c2) {   // S = 1024 = 256 chunks of K=4
      const int kk = c2 * 4 + khi;
      v2f av = *(const v2f*)(&sc[mrow * WIN + kk]);        // ds_load_b64
      v2f bv;
      bv.x = vp[(size_t)kk       * (NH * EDIM)];           // values[b, kk,   h, d]
      bv.y = vp[(size_t)(kk + 1) * (NH * EDIM)];           // values[b, kk+1, h, d]
      acc = __builtin_amdgcn_wmma_f32_16x16x4_f32(false, av, false, bv,
                                                  (short)0, acc, false, false);
    }
#pragma unroll
    for (int r = 0; r < 8; ++r) {
      int row = r + ((lane >> 4) << 3);
      int l = l0 + row;
      __builtin_nontemporal_store(
          acc[r], &outV[(((size_t)b * WIN + l) * NH + h) * EDIM + d0 + mrow]);
    }
  }
}

extern "C" void kernel_launch(void* const* d_in, const int* in_sizes, int n_in,
                              void* d_out, int out_size, void* d_ws, size_t ws_size,
                              hipStream_t stream) {
  const float* q     = (const float*)d_in[0];
  const float* k     = (const float*)d_in[1];
  const float* v     = (const float*)d_in[2];
  const float* sigma = (const float*)d_in[3];

  float* out       = (float*)d_out;
  float* outV      = out;                                       // [B,L,H,D]
  float* outSeries = outV + (size_t)NB * WIN * NH * EDIM;       // [B,H,L,S]
  float* outPrior  = outSeries + (size_t)NB * NH * WIN * WIN;   // [B,H,L,S]
  float* outSig    = outPrior  + (size_t)NB * NH * WIN * WIN;   // [B,H,L,S]

  dim3 grid(WIN / 16, NH, NB);           // 64 x 8 x 4 = 2048 workgroups
  anomaly_attn_kernel<<<grid, 128, 16 * WIN * sizeof(float), stream>>>(
      q, k, v, sigma, outV, outSeries, outPrior, outSig);
}